// ResGnn_50087908606719
// MI455X (gfx1250) — compile-verified
//
#include <hip/hip_runtime.h>

#define N_NODES 100000
#define N_EDGES 600000
#define DIM     128
#define NLAYERS 4
#define BN_EPS  1e-5f

typedef float v2f __attribute__((ext_vector_type(2)));
typedef float v8f __attribute__((ext_vector_type(8)));

// ---------------------------------------------------------------------------
// Kernel 1: agg = (1+eps)*h ; block 0 zeroes BN stat accumulators (256 floats)
// ---------------------------------------------------------------------------
__global__ void init_agg_kernel(const float* __restrict__ h,
                                const float* __restrict__ eps_l,
                                float* __restrict__ agg,
                                float* __restrict__ stats) {
    if (blockIdx.x == 0 && threadIdx.x < 256) stats[threadIdx.x] = 0.0f;
    int idx = blockIdx.x * blockDim.x + threadIdx.x;
    const int n4 = N_NODES * DIM / 4;
    if (idx < n4) {
        float ev = 1.0f + eps_l[0];
        float4 v = ((const float4*)h)[idx];
        v.x *= ev; v.y *= ev; v.z *= ev; v.w *= ev;
        ((float4*)agg)[idx] = v;
    }
}

// ---------------------------------------------------------------------------
// Kernel 2: per-edge message + scatter-add.  One wave32 per edge; each lane
// handles 4 consecutive features (float4 gather, 4x f32 global atomics).
// ---------------------------------------------------------------------------
__global__ void scatter_kernel(const float* __restrict__ h,
                               const int*   __restrict__ src,
                               const int*   __restrict__ dst,
                               const float* __restrict__ ea,
                               const float* __restrict__ We,
                               const float* __restrict__ be,
                               float* __restrict__ agg) {
    int gt   = blockIdx.x * blockDim.x + threadIdx.x;
    int e    = gt >> 5;               // wave-uniform
    int lane = threadIdx.x & 31;
    if (e >= N_EDGES) return;
    int   s = src[e];
    int   d = dst[e];
    float a = ea[e];
    int   c = lane * 4;
    float4 hv = *(const float4*)(h  + (size_t)s * DIM + c);
    float4 wv = *(const float4*)(We + c);
    float4 bv = *(const float4*)(be + c);
    float m0 = fmaxf(fmaf(a, wv.x, hv.x) + bv.x, 0.0f);
    float m1 = fmaxf(fmaf(a, wv.y, hv.y) + bv.y, 0.0f);
    float m2 = fmaxf(fmaf(a, wv.z, hv.z) + bv.z, 0.0f);
    float m3 = fmaxf(fmaf(a, wv.w, hv.w) + bv.w, 0.0f);
    float* p = agg + (size_t)d * DIM + c;
    atomicAdd(p + 0, m0);
    atomicAdd(p + 1, m1);
    atomicAdd(p + 2, m2);
    atomicAdd(p + 3, m3);
}

// ---------------------------------------------------------------------------
// Kernel 3: Z = A @ W + bias  via V_WMMA_F32_16X16X4_F32.
// One wave per 16-row strip; 8 column tiles (full D=128) held in 8 v8f accs.
// A frag (16x4 f32, 2 VGPR): lanes 0-15 -> K=k0,k0+1 ; lanes 16-31 -> K=k0+2,k0+3
// C/D (16x16 f32, 8 VGPR):   reg r, lane half -> row M = r + 8*half, col = lane&15
// N_NODES = 6250 * 16 exactly -> no partial tiles, EXEC stays all-ones.
// ---------------------------------------------------------------------------
__global__ void gemm1_wmma_kernel(const float* __restrict__ A,
                                  const float* __restrict__ W,
                                  const float* __restrict__ bias,
                                  float* __restrict__ Z) {
    int wid = (blockIdx.x * blockDim.x + threadIdx.x) >> 5;   // wave-uniform
    if (wid >= N_NODES / 16) return;
    int lane = threadIdx.x & 31;
    int half = lane >> 4;
    int l15  = lane & 15;
    int rowbase = wid * 16;
    int arow    = rowbase + l15;

    v8f acc[8];
#pragma unroll
    for (int j = 0; j < 8; ++j) acc[j] = (v8f)0.0f;

    for (int kc = 0; kc < 32; ++kc) {
        int k0 = kc * 4 + half * 2;
        v2f a;
        a.x = A[(size_t)arow * DIM + k0];
        a.y = A[(size_t)arow * DIM + k0 + 1];
#pragma unroll
        for (int j = 0; j < 8; ++j) {
            int col = j * 16 + l15;
            v2f b;
            b.x = W[(size_t)k0 * DIM + col];
            b.y = W[(size_t)(k0 + 1) * DIM + col];
            acc[j] = __builtin_amdgcn_wmma_f32_16x16x4_f32(
                false, a, false, b, (short)0, acc[j], false, false);
        }
    }
#pragma unroll
    for (int j = 0; j < 8; ++j) {
        int col = j * 16 + l15;
        float bv = bias[col];
#pragma unroll
        for (int r = 0; r < 8; ++r) {
            int row = rowbase + r + half * 8;
            Z[(size_t)row * DIM + col] = acc[j][r] + bv;
        }
    }
}

// ---------------------------------------------------------------------------
// Kernel 4: per-column sum / sum-of-squares of Z via LDS f32 atomics.
// stats[0..127]=sum, stats[128..255]=sumsq.
// ---------------------------------------------------------------------------
__global__ void stats_kernel(const float* __restrict__ Z,
                             float* __restrict__ stats) {
    __shared__ float ssum[DIM];
    __shared__ float ssq[DIM];
    int tid = threadIdx.x;
    if (tid < DIM) { ssum[tid] = 0.0f; ssq[tid] = 0.0f; }
    __syncthreads();
    const int n4 = N_NODES * DIM / 4;
    for (int i = blockIdx.x * blockDim.x + tid; i < n4; i += gridDim.x * blockDim.x) {
        float4 v = ((const float4*)Z)[i];
        int col = (i * 4) & (DIM - 1);
        atomicAdd(&ssum[col + 0], v.x);  atomicAdd(&ssq[col + 0], v.x * v.x);
        atomicAdd(&ssum[col + 1], v.y);  atomicAdd(&ssq[col + 1], v.y * v.y);
        atomicAdd(&ssum[col + 2], v.z);  atomicAdd(&ssq[col + 2], v.z * v.z);
        atomicAdd(&ssum[col + 3], v.w);  atomicAdd(&ssq[col + 3], v.w * v.w);
    }
    __syncthreads();
    if (tid < DIM)            atomicAdd(&stats[tid], ssum[tid]);
    else if (tid < 2 * DIM)   atomicAdd(&stats[tid], ssq[tid - DIM]);
}

// ---------------------------------------------------------------------------
// Kernel 5: fold BN (training stats) into per-feature affine scale/shift.
// stats[256..383]=scale, stats[384..511]=shift.
// ---------------------------------------------------------------------------
__global__ void bnprep_kernel(float* __restrict__ stats,
                              const float* __restrict__ gamma,
                              const float* __restrict__ beta) {
    int c = threadIdx.x;
    if (c >= DIM) return;
    float inv_n = 1.0f / (float)N_NODES;
    float mu  = stats[c] * inv_n;
    float var = stats[DIM + c] * inv_n - mu * mu;
    float rs  = rsqrtf(var + BN_EPS);
    float sc  = rs * gamma[c];
    stats[2 * DIM + c] = sc;
    stats[3 * DIM + c] = beta[c] - mu * sc;
}

// ---------------------------------------------------------------------------
// Kernel 6: H_out = [residual? H_in : 0] + relu( relu(Z*scale+shift) @ W2 + b2 )
// BN+ReLU applied while loading the A fragment; WMMA f32 as in gemm1.
// In-place on H (each row touched by exactly one wave) is safe.
// ---------------------------------------------------------------------------
__global__ void gemm2_wmma_kernel(const float* __restrict__ Z,
                                  const float* __restrict__ stats,
                                  const float* __restrict__ W,
                                  const float* __restrict__ bias,
                                  const float* __restrict__ Hin,
                                  float* __restrict__ Hout,
                                  int residual) {
    int wid = (blockIdx.x * blockDim.x + threadIdx.x) >> 5;   // wave-uniform
    if (wid >= N_NODES / 16) return;
    int lane = threadIdx.x & 31;
    int half = lane >> 4;
    int l15  = lane & 15;
    int rowbase = wid * 16;
    int arow    = rowbase + l15;
    const float* scale = stats + 2 * DIM;
    const float* shift = stats + 3 * DIM;

    v8f acc[8];
#pragma unroll
    for (int j = 0; j < 8; ++j) acc[j] = (v8f)0.0f;

    for (int kc = 0; kc < 32; ++kc) {
        int k0 = kc * 4 + half * 2;
        v2f a;
        a.x = fmaxf(fmaf(Z[(size_t)arow * DIM + k0],     scale[k0],     shift[k0]),     0.0f);
        a.y = fmaxf(fmaf(Z[(size_t)arow * DIM + k0 + 1], scale[k0 + 1], shift[k0 + 1]), 0.0f);
#pragma unroll
        for (int j = 0; j < 8; ++j) {
            int col = j * 16 + l15;
            v2f b;
            b.x = W[(size_t)k0 * DIM + col];
            b.y = W[(size_t)(k0 + 1) * DIM + col];
            acc[j] = __builtin_amdgcn_wmma_f32_16x16x4_f32(
                false, a, false, b, (short)0, acc[j], false, false);
        }
    }
#pragma unroll
    for (int j = 0; j < 8; ++j) {
        int col = j * 16 + l15;
        float bv = bias[col];
#pragma unroll
        for (int r = 0; r < 8; ++r) {
            int row = rowbase + r + half * 8;
            float t = fmaxf(acc[j][r] + bv, 0.0f);   // relu(z @ W2 + b2)
            if (residual) t += Hin[(size_t)row * DIM + col];
            Hout[(size_t)row * DIM + col] = t;
        }
    }
}

// ---------------------------------------------------------------------------
// Host-side launch: 4 GINE layers, 6 kernels each, all on `stream`.
// ---------------------------------------------------------------------------
extern "C" void kernel_launch(void* const* d_in, const int* in_sizes, int n_in,
                              void* d_out, int out_size, void* d_ws, size_t ws_size,
                              hipStream_t stream) {
    const float* x     = (const float*)d_in[0];
    const int*   eidx  = (const int*)  d_in[1];   // [2, E]
    const float* eattr = (const float*)d_in[2];   // [E, 1]
    const float* We    = (const float*)d_in[3];   // [L, 1, D]
    const float* be    = (const float*)d_in[4];   // [L, D]
    const float* eps   = (const float*)d_in[5];   // [L]
    const float* W1    = (const float*)d_in[6];   // [L, D, D]
    const float* b1    = (const float*)d_in[7];   // [L, D]
    const float* gamma = (const float*)d_in[8];   // [L, D]
    const float* beta  = (const float*)d_in[9];   // [L, D]
    const float* W2    = (const float*)d_in[10];  // [L, D, D]
    const float* b2    = (const float*)d_in[11];  // [L, D]
    float* out = (float*)d_out;

    const int* src = eidx;
    const int* dst = eidx + N_EDGES;

    float* agg   = (float*)d_ws;                         // N*D
    float* z1    = agg + (size_t)N_NODES * DIM;          // N*D
    float* stats = z1  + (size_t)N_NODES * DIM;          // 512 floats

    const int initBlocks    = (N_NODES * DIM / 4 + 255) / 256;
    const int scatterBlocks = (N_EDGES * 32 + 255) / 256;
    const int nWaves        = N_NODES / 16;              // 6250 row strips
    const int gemmBlocks    = (nWaves * 32 + 127) / 128; // 4 waves per block

    for (int l = 0; l < NLAYERS; ++l) {
        const float* h_in = (l == 0) ? x : out;
        hipLaunchKernelGGL(init_agg_kernel, dim3(initBlocks), dim3(256), 0, stream,
                           h_in, eps + l, agg, stats);
        hipLaunchKernelGGL(scatter_kernel, dim3(scatterBlocks), dim3(256), 0, stream,
                           h_in, src, dst, eattr, We + (size_t)l * DIM,
                           be + (size_t)l * DIM, agg);
        hipLaunchKernelGGL(gemm1_wmma_kernel, dim3(gemmBlocks), dim3(128), 0, stream,
                           agg, W1 + (size_t)l * DIM * DIM, b1 + (size_t)l * DIM, z1);
        hipLaunchKernelGGL(stats_kernel, dim3(512), dim3(256), 0, stream,
                           z1, stats);
        hipLaunchKernelGGL(bnprep_kernel, dim3(1), dim3(128), 0, stream,
                           stats, gamma + (size_t)l * DIM, beta + (size_t)l * DIM);
        hipLaunchKernelGGL(gemm2_wmma_kernel, dim3(gemmBlocks), dim3(128), 0, stream,
                           z1, stats, W2 + (size_t)l * DIM * DIM, b2 + (size_t)l * DIM,
                           h_in, out, (l == 0) ? 0 : 1);
    }
    (void)in_sizes; (void)n_in; (void)out_size; (void)ws_size;
}